// oMamba_61950608277760
// MI455X (gfx1250) — compile-verified
//
#include <hip/hip_runtime.h>
#include <hip/hip_bf16.h>
#include <math.h>

// Set to 0 if the tensor_load_to_lds inline-asm path fails to assemble.
#ifndef USE_TDM
#define USE_TDM 1
#endif

// ---------------------------------------------------------------------------
// Sizes
// ---------------------------------------------------------------------------
#define CIN     256
#define CMID    512
#define LSEQ    16384
#define DMODEL  12
#define PTOT    (LSEQ * DMODEL)     // 196608 spatial positions
#define TILEP   32                  // spatial positions per block (2 p-tiles)
#define LDSTR   48                  // padded LDS row stride in floats
#define DINNER  24
#define DSTATE  16
#define DTRANK  4
#define NCHUNK  64
#define CHLEN   (LSEQ / NCHUNK)     // 256

typedef float v2f __attribute__((ext_vector_type(2)));
typedef float v8f __attribute__((ext_vector_type(8)));
typedef unsigned int v4u __attribute__((ext_vector_type(4)));
typedef unsigned int v8u __attribute__((ext_vector_type(8)));

// Workspace layout (float offsets)
#define OFF_AWS    0                         // 512*256 swizzled conv1 weights
#define OFF_B1F    131072                    // 512
#define OFF_W2F    131584                    // 512
#define OFF_B2F    132096                    // 16 (1 used)
#define OFF_SEQ    132112                    // 196608
#define OFF_DELTA  328720                    // 16384*24
#define OFF_UC     721936                    // 16384*24
#define OFF_BM     1115152                   // 16384*16
#define OFF_TAIL   1377296                   // 48 (Cm_last[16], res_last[24])
#define OFF_APROD  1377344                   // 64*384
#define OFF_BACC   1401920                   // 64*384
// total ~1.43M floats = ~5.7 MB of d_ws

__device__ __forceinline__ float softplus_f(float v) {
  return (v > 20.0f) ? v : log1pf(expf(v));
}
__device__ __forceinline__ float silu_f(float v) {
  return v / (1.0f + expf(-v));
}

// ---------------------------------------------------------------------------
// K0: fold BN1/BN2 into conv weights; swizzle conv1 weights into WMMA f32
// 16x4 A-operand layout:  Aws[((t*64+s)*32 + lane)] = float2 where
//   o = t*16 + (lane&15),  c = 4*s + 2*(lane>>4) + {0,1}
// ---------------------------------------------------------------------------
__global__ __launch_bounds__(256) void prep_kernel(
    const float* __restrict__ conv1_w, const float* __restrict__ conv1_b,
    const float* __restrict__ g1, const float* __restrict__ be1,
    const float* __restrict__ mu1, const float* __restrict__ va1,
    const float* __restrict__ conv2_w, const float* __restrict__ conv2_b,
    const float* __restrict__ g2, const float* __restrict__ be2,
    const float* __restrict__ mu2, const float* __restrict__ va2,
    float* __restrict__ Aws, float* __restrict__ b1f,
    float* __restrict__ W2f, float* __restrict__ b2f)
{
  const int tid = blockIdx.x * blockDim.x + threadIdx.x;   // 65536 threads
  if (tid < CMID * CIN / 2) {
    const int j  = tid & 31;
    const int s  = (tid >> 5) & 63;
    const int t  = tid >> 11;
    const int o  = t * 16 + (j & 15);
    const int c0 = 4 * s + 2 * (j >> 4);
    const float inv = g1[o] * rsqrtf(va1[o] + 1e-5f);
    Aws[2 * tid + 0] = conv1_w[o * CIN + c0 + 0] * inv;
    Aws[2 * tid + 1] = conv1_w[o * CIN + c0 + 1] * inv;
  }
  if (tid < CMID) {
    const float inv = g1[tid] * rsqrtf(va1[tid] + 1e-5f);
    b1f[tid] = conv1_b[tid] * inv + be1[tid] - mu1[tid] * inv;
    const float inv2 = g2[0] * rsqrtf(va2[0] + 1e-5f);
    W2f[tid] = conv2_w[tid] * inv2;
    if (tid == 0) b2f[0] = conv2_b[0] * inv2 + be2[0] - mu2[0] * inv2;
  }
}

// ---------------------------------------------------------------------------
// K1: fused conv1+BN1+ReLU+conv2+BN2+ReLU via V_WMMA_F32_16X16X4_F32.
// Block = 32 spatial positions x all 512 mid channels; 8 waves x 4 o-tiles
// x 2 p-tiles.  B tile (256x32 f32) staged into LDS by the Tensor Data
// Mover with 16-dword padding every 32 dwords -> LDS row stride 48 floats,
// which makes each ds_load_2addr_b32 hit 64 distinct banks.
// ---------------------------------------------------------------------------
__global__ __launch_bounds__(256) void conv_fused_kernel(
    const float* __restrict__ x, const float* __restrict__ Aws,
    const float* __restrict__ b1f, const float* __restrict__ W2f,
    const float* __restrict__ b2f, float* __restrict__ seq)
{
  __shared__ float Bt[CIN * LDSTR];   // 48 KB padded tile of x: [c][48]
  __shared__ float b1s[CMID];
  __shared__ float w2s[CMID];
  __shared__ float partial[TILEP];

  const int tid = threadIdx.x;
  const int p0  = blockIdx.x * TILEP;

  b1s[tid]       = b1f[tid];
  b1s[tid + 256] = b1f[tid + 256];
  w2s[tid]       = W2f[tid];
  w2s[tid + 256] = W2f[tid + 256];
  if (tid < TILEP) partial[tid] = 0.0f;

#if USE_TDM
  // ---- Tensor Data Mover: 2D tile, 32 floats x 256 rows, LDS-padded ----
  if (tid < 32) {
    const unsigned long long ga = (unsigned long long)(uintptr_t)(x + p0);
    const unsigned lds_off = (unsigned)(uintptr_t)(&Bt[0]);  // LDS byte addr

    v4u g0;
    g0.x = 1u;                                  // count=1, user descriptor
    g0.y = lds_off;                             // lds_addr
    g0.z = (unsigned)(ga & 0xFFFFFFFFull);      // global_addr[31:0]
    g0.w = (unsigned)((ga >> 32) & 0x01FFFFFFull) | 0x80000000u; // type=2

    v8u g1;
    // data_size=4B (2<<16) | pad_enable (1<<20) | pad_interval=32dw (4<<22)
    // | pad_amount=16dw (15<<25)
    g1.s0 = (2u << 16) | (1u << 20) | (4u << 22) | (15u << 25);
    g1.s1 = (unsigned)TILEP << 16;              // tensor_dim0[15:0] @bits63:48
    g1.s2 = (unsigned)CIN << 16;    // tensor_dim0[31:16]=0, tensor_dim1[15:0]
    g1.s3 = (unsigned)TILEP << 16;  // tensor_dim1[31:16]=0, tile_dim0=32
    g1.s4 = (unsigned)CIN;          // tile_dim1=256, tile_dim2=0
    g1.s5 = (unsigned)PTOT;         // tensor_dim0_stride[31:0] = 196608
    g1.s6 = 0u;                     // stride[47:32]=0, dim1_stride lo
    g1.s7 = 0u;
    asm volatile("tensor_load_to_lds %0, %1" : : "s"(g0), "s"(g1) : "memory");
    __builtin_amdgcn_s_wait_tensorcnt(0);
  }
#else
  // ---- fallback: cooperative float4 loads into the padded layout ----
  #pragma unroll
  for (int it = 0; it < 8; ++it) {
    const int idx = tid + it * 256;        // 0..2047
    const int c   = idx >> 3;
    const int sg  = (idx & 7) << 2;        // 0,4,...,28
    const float4 v =
        *reinterpret_cast<const float4*>(x + (size_t)c * PTOT + p0 + sg);
    *reinterpret_cast<float4*>(&Bt[c * LDSTR + sg]) = v;
  }
#endif
  __syncthreads();

  const int wave = tid >> 5;
  const int lane = tid & 31;
  const int half = lane >> 4;   // which 16-lane half
  const int col  = lane & 15;   // N (spatial) index within a p-tile

  v8f acc00 = {}, acc10 = {}, acc20 = {}, acc30 = {};   // p-tile 0
  v8f acc01 = {}, acc11 = {}, acc21 = {}, acc31 = {};   // p-tile 1
  const int t0 = wave * 4;      // first o-tile for this wave
  const v2f* __restrict__ A2 = reinterpret_cast<const v2f*>(Aws);

  #pragma unroll 4
  for (int s = 0; s < 64; ++s) {
    // B operand: f32 4x16; VGPR0 holds rows {4s, 4s+2}, VGPR1 rows {4s+1,4s+3}
    const int r = 4 * s + 2 * half;
    v2f b0, b1;
    b0.x = Bt[r * LDSTR + col];
    b0.y = Bt[(r + 1) * LDSTR + col];
    b1.x = Bt[r * LDSTR + 16 + col];
    b1.y = Bt[(r + 1) * LDSTR + 16 + col];

    const v2f a0 = A2[((size_t)(t0 + 0) * 64 + s) * 32 + lane];
    const v2f a1 = A2[((size_t)(t0 + 1) * 64 + s) * 32 + lane];
    const v2f a2 = A2[((size_t)(t0 + 2) * 64 + s) * 32 + lane];
    const v2f a3 = A2[((size_t)(t0 + 3) * 64 + s) * 32 + lane];

    acc00 = __builtin_amdgcn_wmma_f32_16x16x4_f32(false, a0, false, b0,
                                                  (short)0, acc00, false, false);
    acc01 = __builtin_amdgcn_wmma_f32_16x16x4_f32(false, a0, false, b1,
                                                  (short)0, acc01, false, false);
    acc10 = __builtin_amdgcn_wmma_f32_16x16x4_f32(false, a1, false, b0,
                                                  (short)0, acc10, false, false);
    acc11 = __builtin_amdgcn_wmma_f32_16x16x4_f32(false, a1, false, b1,
                                                  (short)0, acc11, false, false);
    acc20 = __builtin_amdgcn_wmma_f32_16x16x4_f32(false, a2, false, b0,
                                                  (short)0, acc20, false, false);
    acc21 = __builtin_amdgcn_wmma_f32_16x16x4_f32(false, a2, false, b1,
                                                  (short)0, acc21, false, false);
    acc30 = __builtin_amdgcn_wmma_f32_16x16x4_f32(false, a3, false, b0,
                                                  (short)0, acc30, false, false);
    acc31 = __builtin_amdgcn_wmma_f32_16x16x4_f32(false, a3, false, b1,
                                                  (short)0, acc31, false, false);
  }

  // Epilogue: h1 = relu(acc + b1f), reduce over mid channels with W2f.
  // C/D layout: acc[v], lane -> row M = v + 8*half, col N = lane&15.
  const int mrow = 8 * half;
  float lsum0 = 0.0f, lsum1 = 0.0f;
  #pragma unroll
  for (int v = 0; v < 8; ++v) {
    const int o0 = (t0 + 0) * 16 + v + mrow;
    const int o1 = (t0 + 1) * 16 + v + mrow;
    const int o2 = (t0 + 2) * 16 + v + mrow;
    const int o3 = (t0 + 3) * 16 + v + mrow;
    lsum0 += w2s[o0] * fmaxf(acc00[v] + b1s[o0], 0.0f);
    lsum0 += w2s[o1] * fmaxf(acc10[v] + b1s[o1], 0.0f);
    lsum0 += w2s[o2] * fmaxf(acc20[v] + b1s[o2], 0.0f);
    lsum0 += w2s[o3] * fmaxf(acc30[v] + b1s[o3], 0.0f);
    lsum1 += w2s[o0] * fmaxf(acc01[v] + b1s[o0], 0.0f);
    lsum1 += w2s[o1] * fmaxf(acc11[v] + b1s[o1], 0.0f);
    lsum1 += w2s[o2] * fmaxf(acc21[v] + b1s[o2], 0.0f);
    lsum1 += w2s[o3] * fmaxf(acc31[v] + b1s[o3], 0.0f);
  }
  lsum0 += __shfl_xor(lsum0, 16, 32);        // combine the two M-halves
  lsum1 += __shfl_xor(lsum1, 16, 32);
  if (half == 0) {
    atomicAdd(&partial[col], lsum0);         // ds_add_f32 across waves
    atomicAdd(&partial[16 + col], lsum1);
  }
  __syncthreads();

  if (tid < TILEP) {
    seq[p0 + tid] = fmaxf(partial[tid] + b2f[0], 0.0f);
  }
}

// ---------------------------------------------------------------------------
// K2a: per-timestep Mamba preprocessing. One thread per l.
// ---------------------------------------------------------------------------
__global__ __launch_bounds__(256) void mamba_prep_kernel(
    const float* __restrict__ seq, const float* __restrict__ in_proj_w,
    const float* __restrict__ conv1d_w, const float* __restrict__ conv1d_b,
    const float* __restrict__ x_proj_w, const float* __restrict__ dt_proj_w,
    const float* __restrict__ dt_proj_b,
    float* __restrict__ delta_o, float* __restrict__ uc_o,
    float* __restrict__ Bm_o, float* __restrict__ tail_o)
{
  const int l = blockIdx.x * blockDim.x + threadIdx.x;
  if (l >= LSEQ) return;

  float uc[DINNER];
  #pragma unroll
  for (int d = 0; d < DINNER; ++d) {
    float acc = conv1d_b[d];
    #pragma unroll
    for (int k = 0; k < 4; ++k) {
      const int j = l - 3 + k;
      if (j >= 0) {
        float u = 0.0f;
        #pragma unroll
        for (int m = 0; m < DMODEL; ++m)
          u += in_proj_w[d * DMODEL + m] * seq[j * DMODEL + m];
        acc += conv1d_w[d * 4 + k] * u;
      }
    }
    uc[d] = silu_f(acc);
  }

  float xd4[DTRANK];
  for (int r = 0; r < DTRANK + 2 * DSTATE; ++r) {
    float v = 0.0f;
    #pragma unroll
    for (int d = 0; d < DINNER; ++d) v += x_proj_w[r * DINNER + d] * uc[d];
    if (r < DTRANK) {
      xd4[r] = v;
    } else if (r < DTRANK + DSTATE) {
      Bm_o[l * DSTATE + (r - DTRANK)] = v;
    } else if (l == LSEQ - 1) {
      tail_o[r - DTRANK - DSTATE] = v;            // Cm_last[0..15]
    }
  }

  #pragma unroll
  for (int d = 0; d < DINNER; ++d) {
    float dt = dt_proj_b[d];
    #pragma unroll
    for (int r = 0; r < DTRANK; ++r) dt += dt_proj_w[d * DTRANK + r] * xd4[r];
    delta_o[l * DINNER + d] = softplus_f(dt);
    uc_o[l * DINNER + d]    = uc[d];
  }

  if (l == LSEQ - 1) {
    #pragma unroll
    for (int d = 0; d < DINNER; ++d) {
      float r = 0.0f;
      #pragma unroll
      for (int m = 0; m < DMODEL; ++m)
        r += in_proj_w[(DINNER + d) * DMODEL + m] * seq[l * DMODEL + m];
      tail_o[DSTATE + d] = r;                     // res_last[0..23]
    }
  }
}

// ---------------------------------------------------------------------------
// K2c: chunked linear-recurrence scan. h_l = a_l*h_{l-1} + b_l per (d,n).
// ---------------------------------------------------------------------------
__global__ __launch_bounds__(384) void scan_chunk_kernel(
    const float* __restrict__ delta, const float* __restrict__ ucv,
    const float* __restrict__ Bmv, const float* __restrict__ A_log,
    float* __restrict__ Aprod, float* __restrict__ Bacc)
{
  const int p  = threadIdx.x;        // 0..383
  const int d  = p >> 4;
  const int n  = p & 15;
  const int ch = blockIdx.x;         // 0..63
  const float Aval = -expf(A_log[d * DSTATE + n]);

  float ar = 1.0f, br = 0.0f;
  const int l0 = ch * CHLEN;
  for (int t = 0; t < CHLEN; ++t) {
    const int l    = l0 + t;
    const float dl = delta[l * DINNER + d];
    const float a  = expf(dl * Aval);
    const float b  = dl * Bmv[l * DSTATE + n] * ucv[l * DINNER + d];
    br = a * br + b;
    ar *= a;
  }
  Aprod[ch * 384 + p] = ar;
  Bacc[ch * 384 + p]  = br;
}

// ---------------------------------------------------------------------------
// K2d: combine chunks sequentially, project last-step output to the scalar.
// ---------------------------------------------------------------------------
__global__ __launch_bounds__(384) void finalize_kernel(
    const float* __restrict__ Aprod, const float* __restrict__ Bacc,
    const float* __restrict__ tail, const float* __restrict__ ucv,
    const float* __restrict__ Dp, const float* __restrict__ out_proj_w,
    const float* __restrict__ lin_w, const float* __restrict__ lin_b,
    float* __restrict__ out)
{
  __shared__ float hc[384];
  __shared__ float yl[DINNER];
  __shared__ float ol[DMODEL];

  const int p = threadIdx.x;
  const int n = p & 15;
  float h = 0.0f;
  for (int ch = 0; ch < NCHUNK; ++ch)
    h = Aprod[ch * 384 + p] * h + Bacc[ch * 384 + p];
  hc[p] = h * tail[n];                       // h[d][n] * Cm_last[n]
  __syncthreads();

  if (p < DINNER) {
    float y = 0.0f;
    for (int n2 = 0; n2 < DSTATE; ++n2) y += hc[p * DSTATE + n2];
    y += ucv[(LSEQ - 1) * DINNER + p] * Dp[p];
    y *= silu_f(tail[DSTATE + p]);           // gate by silu(res_last)
    yl[p] = y;
  }
  __syncthreads();
  if (p < DMODEL) {
    float o = 0.0f;
    for (int d2 = 0; d2 < DINNER; ++d2) o += out_proj_w[p * DINNER + d2] * yl[d2];
    ol[p] = o;
  }
  __syncthreads();
  if (p == 0) {
    float s = lin_b[0];
    for (int m = 0; m < DMODEL; ++m) s += lin_w[m] * ol[m];
    out[0] = s;
  }
}

// ---------------------------------------------------------------------------
extern "C" void kernel_launch(void* const* d_in, const int* in_sizes, int n_in,
                              void* d_out, int out_size, void* d_ws,
                              size_t ws_size, hipStream_t stream) {
  const float* x         = (const float*)d_in[0];
  const float* conv1_w   = (const float*)d_in[1];
  const float* conv1_b   = (const float*)d_in[2];
  const float* bn1_gamma = (const float*)d_in[3];
  const float* bn1_beta  = (const float*)d_in[4];
  const float* bn1_mean  = (const float*)d_in[5];
  const float* bn1_var   = (const float*)d_in[6];
  const float* conv2_w   = (const float*)d_in[7];
  const float* conv2_b   = (const float*)d_in[8];
  const float* bn2_gamma = (const float*)d_in[9];
  const float* bn2_beta  = (const float*)d_in[10];
  const float* bn2_mean  = (const float*)d_in[11];
  const float* bn2_var   = (const float*)d_in[12];
  const float* in_proj_w = (const float*)d_in[13];
  const float* conv1d_w  = (const float*)d_in[14];
  const float* conv1d_b  = (const float*)d_in[15];
  const float* x_proj_w  = (const float*)d_in[16];
  const float* dt_proj_w = (const float*)d_in[17];
  const float* dt_proj_b = (const float*)d_in[18];
  const float* A_log     = (const float*)d_in[19];
  const float* Dp        = (const float*)d_in[20];
  const float* out_proj_w= (const float*)d_in[21];
  const float* lin_w     = (const float*)d_in[22];
  const float* lin_b     = (const float*)d_in[23];

  float* ws  = (float*)d_ws;
  float* out = (float*)d_out;

  prep_kernel<<<dim3((CMID * CIN / 2) / 256), 256, 0, stream>>>(
      conv1_w, conv1_b, bn1_gamma, bn1_beta, bn1_mean, bn1_var,
      conv2_w, conv2_b, bn2_gamma, bn2_beta, bn2_mean, bn2_var,
      ws + OFF_AWS, ws + OFF_B1F, ws + OFF_W2F, ws + OFF_B2F);

  conv_fused_kernel<<<dim3(PTOT / TILEP), 256, 0, stream>>>(
      x, ws + OFF_AWS, ws + OFF_B1F, ws + OFF_W2F, ws + OFF_B2F,
      ws + OFF_SEQ);

  mamba_prep_kernel<<<dim3(LSEQ / 256), 256, 0, stream>>>(
      ws + OFF_SEQ, in_proj_w, conv1d_w, conv1d_b, x_proj_w, dt_proj_w,
      dt_proj_b, ws + OFF_DELTA, ws + OFF_UC, ws + OFF_BM, ws + OFF_TAIL);

  scan_chunk_kernel<<<dim3(NCHUNK), 384, 0, stream>>>(
      ws + OFF_DELTA, ws + OFF_UC, ws + OFF_BM, A_log,
      ws + OFF_APROD, ws + OFF_BACC);

  finalize_kernel<<<dim3(1), 384, 0, stream>>>(
      ws + OFF_APROD, ws + OFF_BACC, ws + OFF_TAIL, ws + OFF_UC, Dp,
      out_proj_w, lin_w, lin_b, out);
}